// GraphSAGEModel_29901562315009
// MI455X (gfx1250) — compile-verified
//
#include <hip/hip_runtime.h>
#include <hip/hip_bf16.h>

typedef __attribute__((ext_vector_type(2))) float v2f;
typedef __attribute__((ext_vector_type(8))) float v8f;

#define D 64

// ---------------------------------------------------------------------------
// Edge aggregation: neigh[dst] += w * h[src]   (one wave32 per edge, lane
// handles a float2 chunk -> each edge touches exactly one 256B row, coalesced).
// Feature table (25.6 MB) is L2-resident (192 MB L2); atomics are hardware
// global_atomic_add_f32 (no-return) via unsafeAtomicAdd.
// ---------------------------------------------------------------------------
__global__ __launch_bounds__(256) void sage_edge_agg(
    const float* __restrict__ h, const int* __restrict__ src,
    const int* __restrict__ dst, const float* __restrict__ w,
    float* __restrict__ neigh, int n_edges)
{
    long long tid = (long long)blockIdx.x * blockDim.x + threadIdx.x;
    int e = (int)(tid >> 5);
    if (e >= n_edges) return;
    int lane = (int)(tid & 31);

    int s = src[e];
    int d = dst[e];
    float we = w[e];

    v2f x = *(const v2f*)&h[(size_t)s * D + lane * 2];
    float* p = &neigh[(size_t)d * D + lane * 2];
    unsafeAtomicAdd(p + 0, x.x * we);
    unsafeAtomicAdd(p + 1, x.y * we);
}

// ---------------------------------------------------------------------------
// Fused dual-GEMM:  out = act( Hself @ Wself + Hneigh @ Wneigh + bias )
// One wave computes a 16-row x 64-col output strip with V_WMMA_F32_16X16X4_F32.
// Weights staged in LDS *pre-packed as K-pair float2* so each B fragment is a
// single aligned ds_load_b64 straight into an even VGPR pair (no mov shuffles).
// Safe when out == Hself (each wave reads only its own rows, and all loads
// feed the accumulators before any store).
// ---------------------------------------------------------------------------
__global__ __launch_bounds__(256) void sage_gemm(
    const float* Hself, const float* __restrict__ Hneigh,
    const float* __restrict__ Wself, const float* __restrict__ Wneigh,
    const float* __restrict__ bias, float* out,
    int n_nodes, int do_relu)
{
    // Packed layout: sW[p*D + col] = { W[2p][col], W[2p+1][col] },  p = 0..31
    __shared__ v2f sWs[(D / 2) * D];
    __shared__ v2f sWn[(D / 2) * D];

    const int t = threadIdx.x;
    // 256 threads x 8 entries = 2048 float2 per matrix.
    for (int i = t; i < (D / 2) * D; i += 256) {
        const int p   = i >> 6;        // K-pair index
        const int col = i & 63;
        v2f ws, wn;
        ws.x = Wself[(2 * p + 0) * D + col];
        ws.y = Wself[(2 * p + 1) * D + col];
        wn.x = Wneigh[(2 * p + 0) * D + col];
        wn.y = Wneigh[(2 * p + 1) * D + col];
        sWs[i] = ws;
        sWn[i] = wn;
    }
    __syncthreads();

    const int wave  = t >> 5;
    const int lane  = t & 31;
    const int strip = blockIdx.x * 8 + wave;          // 16-row strip
    if (strip * 16 >= n_nodes) return;                // wave-uniform exit (EXEC all-1 for WMMA)
    const int row0  = strip * 16;

    const int r     = lane & 15;                      // row (A) / col (B,C,D)
    const int khalf = lane >> 4;                      // K-half selector

    // Accumulators: 4 N-blocks of 16 cols. Init with bias (bias is per-column).
    v8f acc[4];
#pragma unroll
    for (int nb = 0; nb < 4; ++nb) {
        float bv = bias[nb * 16 + r];
#pragma unroll
        for (int v = 0; v < 8; ++v) acc[nb][v] = bv;
    }

    const float* hs = Hself  + (size_t)row0 * D;
    const float* hn = Hneigh + (size_t)row0 * D;

#pragma unroll 4
    for (int k = 0; k < 16; ++k) {                    // K = 64 in steps of 4
        const int kk = k * 4 + khalf * 2;             // lane's K columns
        const int kp = 2 * k + khalf;                 // packed K-pair row
        // A fragments (16x4): lane holds A[r][kk], A[r][kk+1]
        v2f aS = *(const v2f*)&hs[r * D + kk];
        v2f aN = *(const v2f*)&hn[r * D + kk];
#pragma unroll
        for (int nb = 0; nb < 4; ++nb) {
            const int col = nb * 16 + r;
            v2f bS = sWs[kp * D + col];               // one ds_load_b64 each
            v2f bN = sWn[kp * D + col];
            acc[nb] = __builtin_amdgcn_wmma_f32_16x16x4_f32(
                false, aS, false, bS, (short)0, acc[nb], false, false);
            acc[nb] = __builtin_amdgcn_wmma_f32_16x16x4_f32(
                false, aN, false, bN, (short)0, acc[nb], false, false);
        }
    }

    if (do_relu) {
#pragma unroll
        for (int nb = 0; nb < 4; ++nb)
#pragma unroll
            for (int v = 0; v < 8; ++v) acc[nb][v] = fmaxf(acc[nb][v], 0.0f);
    }

    // Store: C/D layout -> lane writes rows (v + 8*khalf), col nb*16 + r.
#pragma unroll
    for (int nb = 0; nb < 4; ++nb) {
        const int col = nb * 16 + r;
#pragma unroll
        for (int v = 0; v < 8; ++v) {
            out[(size_t)(row0 + v + 8 * khalf) * D + col] = acc[nb][v];
        }
    }
}

// ---------------------------------------------------------------------------
extern "C" void kernel_launch(void* const* d_in, const int* in_sizes, int n_in,
                              void* d_out, int out_size, void* d_ws, size_t ws_size,
                              hipStream_t stream) {
    const float* features = (const float*)d_in[0];
    const int*   esrc     = (const int*)d_in[1];
    const int*   edst     = (const int*)d_in[2];
    const float* ew       = (const float*)d_in[3];
    const float* Ws0      = (const float*)d_in[4];
    const float* Wn0      = (const float*)d_in[5];
    const float* b0       = (const float*)d_in[6];
    const float* Ws1      = (const float*)d_in[7];
    const float* Wn1      = (const float*)d_in[8];
    const float* b1       = (const float*)d_in[9];

    float* out   = (float*)d_out;
    float* neigh = (float*)d_ws;          // N x 64 f32 accumulation buffer (25.6 MB)

    const int N = in_sizes[0] / D;        // 100000
    const int E = in_sizes[1];            // 3200000
    const size_t nbytes = (size_t)N * D * sizeof(float);

    const int aggBlocks  = (int)(((long long)E * 32 + 255) / 256);
    const int strips     = (N + 15) / 16;
    const int gemmBlocks = (strips + 7) / 8;

    // ---- Layer 0 ----
    hipMemsetAsync(neigh, 0, nbytes, stream);
    sage_edge_agg<<<aggBlocks, 256, 0, stream>>>(features, esrc, edst, ew, neigh, E);
    sage_gemm<<<gemmBlocks, 256, 0, stream>>>(features, neigh, Ws0, Wn0, b0, out, N, 1);

    // ---- Layer 1 (h0 lives in d_out; final GEMM is row-local in-place) ----
    hipMemsetAsync(neigh, 0, nbytes, stream);
    sage_edge_agg<<<aggBlocks, 256, 0, stream>>>(out, esrc, edst, ew, neigh, E);
    sage_gemm<<<gemmBlocks, 256, 0, stream>>>(out, neigh, Ws1, Wn1, b1, out, N, 0);
}